// UniformAffineQuantizer_42700564857200
// MI455X (gfx1250) — compile-verified
//
#include <hip/hip_runtime.h>
#include <stdint.h>

// Problem geometry (fixed by the reference)
#define ROWS 4096
#define COLS 11008
#define GRP  128
#define GPR  (COLS / GRP)        // 86 groups per row
#define TPB  256
#define F4_PER_ROW (COLS / 4)    // 2752 float4 per row
#define NIT ((F4_PER_ROW + TPB - 1) / TPB)  // 11 tiles per row

typedef float vf4 __attribute__((ext_vector_type(4)));
typedef int   vi4 __attribute__((ext_vector_type(4)));

// one dequant step using the folded per-group constants
// e = { inv_s, s, -rzp, qmaxf - rzp }
__device__ __forceinline__ float fq1(float xv, vf4 e) {
    float t = __builtin_rintf(xv * e.x);   // v_rndne_f32: round half-to-even (jnp.round)
    t = fmaxf(t, e.z);
    t = fminf(t, e.w);
    return t * e.y;
}

__global__ __launch_bounds__(TPB) void fakequant_rowblock_kernel(
    const float* __restrict__ x,
    const float* __restrict__ scale,
    const float* __restrict__ zero_point,
    const int*   __restrict__ qmax,
    const int*   __restrict__ invcol,
    float*       __restrict__ out)
{
    __shared__ vf4 table[GPR + 2];     // per-group folded constants for this row
    __shared__ vf4 xbuf[2][TPB];       // async-load double buffer (8 KB)

    const int r   = blockIdx.x;
    const int tid = threadIdx.x;

    // ---- Phase 1: build the 86-entry group-parameter table for row r ----
    if (tid < GPR) {
        const int g = r * GPR + tid;
        float s  = scale[g];
        s = fminf(fmaxf(s, 1.0e-4f), 1.0e4f);          // clip(scale)
        const float qf = (float)qmax[g];
        float rz = __builtin_rintf(zero_point[g]);      // round(zero_point)
        rz = fminf(fmaxf(rz, 0.0f), qf);                // clip to [0, qmax]
        vf4 e;
        e.x = 1.0f / s;   // inv_s
        e.y = s;          // s
        e.z = -rz;        // lower clamp after folding
        e.w = qf - rz;    // upper clamp after folding
        table[tid] = e;
    }
    __syncthreads();

    const float* xrow = x   + (size_t)r * COLS;
    float*       orow = out + (size_t)r * COLS;
    const vi4*   ic4  = (const vi4*)invcol;
    const vf4*   x4   = (const vf4*)xrow;
    vf4*         o4   = (vf4*)orow;

    // Per-lane LDS byte addresses: low 32 bits of a generic pointer to LDS
    // are the LDS offset (LDS aperture has zero low bits).
    const uint32_t lds0 = (uint32_t)(uintptr_t)&xbuf[0][tid];
    const uint32_t lds1 = (uint32_t)(uintptr_t)&xbuf[1][tid];

    // ---- Prologue: async-copy tile 0 (gfx1250 ASYNCcnt path) ----
    {
        unsigned long long src = (unsigned long long)(uintptr_t)(x4 + tid); // tid < 2752 always
        asm volatile("global_load_async_to_lds_b128 %0, %1, off"
                     :: "v"(lds0), "v"(src) : "memory");
    }

    // ---- Steady state: 2-deep async pipeline ----
    for (int i = 0; i < NIT; ++i) {
        if (i + 1 < NIT) {
            // Clamp (never EXEC-skip) so every wave issues exactly one async op
            // per tile and s_wait_asynccnt accounting stays exact.
            int c4n = (i + 1) * TPB + tid;
            c4n = (c4n < F4_PER_ROW) ? c4n : (F4_PER_ROW - 1);
            const uint32_t dst = ((i + 1) & 1) ? lds1 : lds0;
            unsigned long long src = (unsigned long long)(uintptr_t)(x4 + c4n);
            asm volatile("global_load_async_to_lds_b128 %0, %1, off"
                         :: "v"(dst), "v"(src) : "memory");
            asm volatile("s_wait_asynccnt 0x1" ::: "memory");  // tile i has landed
        } else {
            asm volatile("s_wait_asynccnt 0x0" ::: "memory");  // drain
        }

        const int c4 = i * TPB + tid;
        if (c4 < F4_PER_ROW) {
            const vf4 xv = xbuf[i & 1][tid];   // ds_load_b128
            const vi4 ic = ic4[c4];            // coalesced, L2-resident
            vf4 o;
            o.x = fq1(xv.x, table[ic.x >> 7]);
            o.y = fq1(xv.y, table[ic.y >> 7]);
            o.z = fq1(xv.z, table[ic.z >> 7]);
            o.w = fq1(xv.w, table[ic.w >> 7]);
            __builtin_nontemporal_store(o, o4 + c4);  // streamed once: NT store
        }
    }
}

extern "C" void kernel_launch(void* const* d_in, const int* in_sizes, int n_in,
                              void* d_out, int out_size, void* d_ws, size_t ws_size,
                              hipStream_t stream) {
    (void)in_sizes; (void)n_in; (void)d_ws; (void)ws_size; (void)out_size;
    // setup_inputs order: x, scale, zero_point, qmax, col_order, invcol
    const float* x      = (const float*)d_in[0];
    const float* scale  = (const float*)d_in[1];
    const float* zp     = (const float*)d_in[2];
    const int*   qmax   = (const int*)d_in[3];
    const int*   invcol = (const int*)d_in[5];   // col_order (d_in[4]) not needed
    float* out = (float*)d_out;

    fakequant_rowblock_kernel<<<dim3(ROWS), dim3(TPB), 0, stream>>>(
        x, scale, zp, qmax, invcol, out);
}